// _LightGCN_1726576856308
// MI455X (gfx1250) — compile-verified
//
#include <hip/hip_runtime.h>
#include <math.h>

#define NU 60000
#define NI 40000
#define NNODE 100000
#define DD 64
#define BB 2048

typedef __attribute__((ext_vector_type(2))) float v2f;
typedef __attribute__((ext_vector_type(8))) float v8f;

static __host__ __device__ inline size_t cdiv(size_t a, size_t b) { return (a + b - 1) / b; }

// ---------------- init: ego = concat(user_emb, item_emb); acc = ego ----------------
__global__ void k_init(const float* __restrict__ ue, const float* __restrict__ ie,
                       float* __restrict__ ego, float* __restrict__ acc) {
  size_t i = (size_t)blockIdx.x * blockDim.x + threadIdx.x;
  if (i >= (size_t)NNODE * DD) return;
  float v = (i < (size_t)NU * DD) ? ue[i] : ie[i - (size_t)NU * DD];
  ego[i] = v;
  acc[i] = v;
}

__global__ void k_zero(float* __restrict__ p, size_t n) {
  size_t i = (size_t)blockIdx.x * blockDim.x + threadIdx.x;
  if (i < n) p[i] = 0.0f;
}

__global__ void k_accadd(float* __restrict__ acc, const float* __restrict__ x, size_t n) {
  size_t i = (size_t)blockIdx.x * blockDim.x + threadIdx.x;
  if (i < n) acc[i] += x[i];
}

// ---------------- SpMM: y[row] += val * x[col], 16 lanes per edge, float4/lane ------
__global__ void k_spmm(const int* __restrict__ idx, const float* __restrict__ val,
                       const float* __restrict__ x, float* __restrict__ y, int nE) {
  size_t t = (size_t)blockIdx.x * blockDim.x + threadIdx.x;
  if (t >= (size_t)nE * 16) return;
  int e = (int)(t >> 4);
  int q = (int)(t & 15);
  int row = idx[e];               // idx[0][e] : scatter target (segment id)
  int col = idx[(size_t)nE + e];  // idx[1][e] : gather source
  float v = val[e];
  const float4 xv = ((const float4*)(x + (size_t)col * DD))[q];
  float* yp = y + (size_t)row * DD + q * 4;
  atomicAdd(yp + 0, v * xv.x);
  atomicAdd(yp + 1, v * xv.y);
  atomicAdd(yp + 2, v * xv.z);
  atomicAdd(yp + 3, v * xv.w);
}

// ---------------- row-wise L2 normalize (wave32 per row, float2 per lane) ----------
__global__ void k_l2norm(const float* __restrict__ x, float* __restrict__ y) {
  int wave = threadIdx.x >> 5;
  int lane = threadIdx.x & 31;
  int row = blockIdx.x * (blockDim.x >> 5) + wave;
  if (row >= NNODE) return;
  const float2 v = ((const float2*)(x + (size_t)row * DD))[lane];
  float s = v.x * v.x + v.y * v.y;
#pragma unroll
  for (int o = 16; o >= 1; o >>= 1) s += __shfl_xor(s, o, 32);
  float inv = 1.0f / fmaxf(sqrtf(s), 1e-12f);
  float2 r; r.x = v.x * inv; r.y = v.y * inv;
  ((float2*)(y + (size_t)row * DD))[lane] = r;
}

// ---------------- sup_logits = (u.i - u.ni) / 16 (1/4 layer-mean folded twice) -----
__global__ void k_sup(const float* __restrict__ acc, const int* __restrict__ users,
                      const int* __restrict__ items, const int* __restrict__ negs,
                      float* __restrict__ out) {
  int b = blockIdx.x * blockDim.x + threadIdx.x;
  if (b >= BB) return;
  const float* pu = acc + (size_t)users[b] * DD;
  const float* pi = acc + (size_t)(NU + items[b]) * DD;
  const float* pn = acc + (size_t)(NU + negs[b]) * DD;
  float s = 0.0f;
#pragma unroll 16
  for (int k = 0; k < DD; ++k) s += pu[k] * (pi[k] - pn[k]);
  out[b] = s * 0.0625f;
}

// ---------------- gather batch rows into compact [B,64] tiles ----------------------
__global__ void k_gather(const float* __restrict__ a1, const float* __restrict__ a2,
                         const int* __restrict__ users, const int* __restrict__ items,
                         float* __restrict__ U1, float* __restrict__ U2,
                         float* __restrict__ I1, float* __restrict__ I2) {
  int t = blockIdx.x * blockDim.x + threadIdx.x;
  if (t >= BB * DD) return;
  int b = t >> 6, k = t & 63;
  size_t ur = (size_t)users[b] * DD + k;
  size_t ir = (size_t)(NU + items[b]) * DD + k;
  U1[t] = a1[ur];
  U2[t] = a2[ur];
  I1[t] = a1[ir];
  I2[t] = a2[ir];
}

__global__ void k_pos(const float* __restrict__ U1, const float* __restrict__ U2,
                      const float* __restrict__ I1, const float* __restrict__ I2,
                      float* __restrict__ pu, float* __restrict__ pi) {
  int b = blockIdx.x * blockDim.x + threadIdx.x;
  if (b >= BB) return;
  float su = 0.0f, si = 0.0f;
  const float* u1 = U1 + (size_t)b * DD;
  const float* u2 = U2 + (size_t)b * DD;
  const float* i1 = I1 + (size_t)b * DD;
  const float* i2 = I2 + (size_t)b * DD;
#pragma unroll 16
  for (int k = 0; k < DD; ++k) { su += u1[k] * u2[k]; si += i1[k] * i2[k]; }
  pu[b] = su;
  pi[b] = si;
}

// ---------------- SSL GEMM: out[m][n] = sum_k A[m][k]*Bt[n][k] - pos[m] ------------
// One wave per 16x16 tile; 16 chained V_WMMA_F32_16X16X4_F32 over K=64.
// A 16x4 layout: lane l<16 -> row l, K pair {k0,k0+1}; lanes 16-31 -> K {k0+2,k0+3}.
// B 4x16 layout mirrors with lane = column. Both are contiguous float2 per lane.
__global__ __launch_bounds__(256) void k_gemm_ssl(
    const float* __restrict__ Am,  // [BB, 64] row-major
    const float* __restrict__ Bt,  // [Nn, 64] row-major (logical B[k][n] = Bt[n][k])
    const float* __restrict__ pos, // [BB]
    float* __restrict__ out, int Nn, int ntiles) {
  const int wave = threadIdx.x >> 5;
  const int lane = threadIdx.x & 31;
  const int ntile = blockIdx.x * 8 + wave;   // wave-uniform guard: EXEC all-ones at WMMA
  if (ntile >= ntiles) return;
  const int m0 = blockIdx.y << 4;
  const int n0 = ntile << 4;
  const int half = lane >> 4;   // 0 -> K pair {0,1}; 1 -> K pair {2,3}
  const int l = lane & 15;

  const float* ap = Am + (size_t)(m0 + l) * DD + 2 * half;
  const float* bp = Bt + (size_t)(n0 + l) * DD + 2 * half;

  v8f c = {0.0f, 0.0f, 0.0f, 0.0f, 0.0f, 0.0f, 0.0f, 0.0f};
#pragma unroll
  for (int k0 = 0; k0 < DD; k0 += 4) {
    v2f a = *(const v2f*)(ap + k0);
    v2f b = *(const v2f*)(bp + k0);
    c = __builtin_amdgcn_wmma_f32_16x16x4_f32(
        /*neg_a=*/false, a, /*neg_b=*/false, b,
        /*c_mod=*/(short)0, c, /*reuse_a=*/false, /*reuse_b=*/false);
  }
  // Epilogue: subtract per-row pos, non-temporal streaming stores (819 MB output).
#pragma unroll
  for (int v = 0; v < 8; ++v) {
    int m = m0 + v + 8 * half;
    size_t o = (size_t)m * (size_t)Nn + (size_t)(n0 + l);
    __builtin_nontemporal_store(c[v] - pos[m], out + o);
  }
}

extern "C" void kernel_launch(void* const* d_in, const int* in_sizes, int n_in,
                              void* d_out, int out_size, void* d_ws, size_t ws_size,
                              hipStream_t stream) {
  const float* user_emb = (const float*)d_in[0];
  const float* item_emb = (const float*)d_in[1];
  const int*   adj_idx  = (const int*)d_in[2];
  const float* adj_val  = (const float*)d_in[3];
  const int*   sg1_idx  = (const int*)d_in[4];
  const float* sg1_val  = (const float*)d_in[5];
  const int*   sg2_idx  = (const int*)d_in[6];
  const float* sg2_val  = (const float*)d_in[7];
  const int*   users    = (const int*)d_in[8];
  const int*   items    = (const int*)d_in[9];
  const int*   negs     = (const int*)d_in[10];
  float* out = (float*)d_out;

  const int E_full = in_sizes[3];  // adj_val count
  const int E_sub  = in_sizes[5];  // sg val count

  const size_t ND = (size_t)NNODE * DD;
  float* ws = (float*)d_ws;
  float* egoA = ws;
  float* egoB = ws + ND;
  float* acc  = ws + 2 * ND;
  float* all1 = ws + 3 * ND;
  float* all2 = ws + 4 * ND;
  float* U1   = ws + 5 * ND;
  float* U2   = U1 + (size_t)BB * DD;
  float* I1   = U2 + (size_t)BB * DD;
  float* I2   = I1 + (size_t)BB * DD;
  float* posu = I2 + (size_t)BB * DD;
  float* posi = posu + BB;

  const int TB = 256;
  const int blkND = (int)cdiv(ND, TB);

  auto run_pass = [&](const int* idx, const float* val, int nE) {
    k_init<<<blkND, TB, 0, stream>>>(user_emb, item_emb, egoA, acc);
    float* xin = egoA;
    float* xout = egoB;
    for (int layer = 0; layer < 3; ++layer) {
      k_zero<<<blkND, TB, 0, stream>>>(xout, ND);
      int blkE = (int)cdiv((size_t)nE * 16, TB);
      k_spmm<<<blkE, TB, 0, stream>>>(idx, val, xin, xout, nE);
      k_accadd<<<blkND, TB, 0, stream>>>(acc, xout, ND);
      float* t = xin; xin = xout; xout = t;
    }
  };

  // Pass 1: full adjacency -> sup_logits (scale 1/16 folded in)
  run_pass(adj_idx, adj_val, E_full);
  k_sup<<<(BB + TB - 1) / TB, TB, 0, stream>>>(acc, users, items, negs, out);

  // Pass 2: subgraph 1 -> normalized table all1
  run_pass(sg1_idx, sg1_val, E_sub);
  k_l2norm<<<NNODE / (TB / 32), TB, 0, stream>>>(acc, all1);

  // Pass 3: subgraph 2 -> normalized table all2
  run_pass(sg2_idx, sg2_val, E_sub);
  k_l2norm<<<NNODE / (TB / 32), TB, 0, stream>>>(acc, all2);

  // Gather batch rows + positive-pair dots
  k_gather<<<(BB * DD + TB - 1) / TB, TB, 0, stream>>>(all1, all2, users, items, U1, U2, I1, I2);
  k_pos<<<(BB + TB - 1) / TB, TB, 0, stream>>>(U1, U2, I1, I2, posu, posi);

  // SSL logits GEMMs (WMMA f32 16x16x4), outputs after sup_logits
  float* outU = out + BB;
  float* outI = outU + (size_t)BB * NU;
  const int utiles = NU / 16;  // 3750
  const int itiles = NI / 16;  // 2500
  dim3 gridU((unsigned)cdiv(utiles, 8), BB / 16);
  dim3 gridI((unsigned)cdiv(itiles, 8), BB / 16);
  k_gemm_ssl<<<gridU, 256, 0, stream>>>(U1, all2, posu, outU, NU, utiles);
  k_gemm_ssl<<<gridI, 256, 0, stream>>>(I1, all2 + (size_t)NU * DD, posi, outI, NI, itiles);
}